// ADTNLinear_41755672052475
// MI455X (gfx1250) — compile-verified
//
#include <hip/hip_runtime.h>
#include <hip/hip_bf16.h>

typedef __attribute__((ext_vector_type(2))) float v2f;
typedef __attribute__((ext_vector_type(4))) float v4f;
typedef __attribute__((ext_vector_type(8))) float v8f;

// ---------------------------------------------------------------------------
// Pass 1: reduce gates (P,4,4,8,8,8,8) -> swizzled W[k] (64x16), laid out so
// the layer kernel's B-fragment loads are coalesced float2 per lane:
//   word = (s*32 + lane)*2 + v  holds  Wmat[K = 4s + v + 2*(lane>=16), N = lane&15]
//   Wmat[q = i*8+j, c] = (c<8) ? G1[k,i,j,c] : G2[k,i,j,c-8]
// ---------------------------------------------------------------------------
__global__ __launch_bounds__(256) void gate_reduce_kernel(
    const float* __restrict__ g, float* __restrict__ W, int P)
{
    int tid = blockIdx.x * 256 + threadIdx.x;
    if (tid >= P * 1024) return;
    int c = tid & 15;          // output column 0..15
    int q = (tid >> 4) & 63;   // i*8+j
    int k = tid >> 10;

    // gates flat: (((k*16 + r)*64 + q)*64 + m*8 + n)
    const float* bk = g + (((size_t)k * 16) * 64 + q) * 64;
    float sum = 0.f;
    if (c < 8) {               // G1[m=c] = sum over r, n
        for (int r = 0; r < 16; ++r) {
            const float* p = bk + (size_t)r * 4096 + c * 8;
            #pragma unroll
            for (int n = 0; n < 8; ++n) sum += p[n];
        }
    } else {                   // G2[n=c-8] = sum over r, m
        for (int r = 0; r < 16; ++r) {
            const float* p = bk + (size_t)r * 4096 + (c - 8);
            #pragma unroll
            for (int m = 0; m < 8; ++m) sum += p[m * 8];
        }
    }
    int s  = q >> 2;
    int v  = q & 1;
    int hi = (q >> 1) & 1;
    int lane = c + 16 * hi;
    int word = (s * 32 + lane) * 2 + v;
    W[(size_t)k * 1024 + word] = sum;
}

// ---------------------------------------------------------------------------
// Pass 2 (x3): per pair k, per 16-batch tile: D(16x16) = O(16x64) x W[k](64x16)
// via 16 chained V_WMMA_F32_16X16X4_F32. MODE 0: gather from x via input_perm;
// MODE 1: h -> h in place; MODE 2: h -> out with alpha*scale + bias epilogue.
// ---------------------------------------------------------------------------
template<int MODE>
__global__ __launch_bounds__(256) void layer_kernel(
    const float* __restrict__ src, float* __restrict__ dst,
    const float* __restrict__ W,   const int* __restrict__ perm,
    const float* __restrict__ pds, const float* __restrict__ bias,
    const float* __restrict__ alpha, int offset)
{
    const int k      = blockIdx.x;
    const int waveId = threadIdx.x >> 5;
    const int lane   = threadIdx.x & 31;
    const int ln     = lane & 15;
    const int hi     = lane >> 4;     // 0: K-cols {4s,4s+1}, 1: {4s+2,4s+3}
    const int c2     = hi * 2;        // x2 column offset for this half-wave
    const int s1     = offset + 2 * k;

    // --- B fragments: whole W[k] (4 KB), coalesced float2 loads -------------
    const float2* wk2 = (const float2*)(W + (size_t)k * 1024);
    v2f Bm[16];
    #pragma unroll
    for (int s = 0; s < 16; ++s) {
        float2 w = wk2[s * 32 + lane];
        Bm[s].x = w.x; Bm[s].y = w.y;
    }

    // --- per-lane epilogue constants (MODE 2): f = s1*8 + N ------------------
    const int fout = s1 * 8 + ln;
    float pd = 0.f, bs = 0.f;
    if (MODE == 2) { pd = alpha[0] * pds[fout]; bs = bias[fout]; }

    // --- permutation indices (MODE 0): block-uniform for x1, lane-half x2 ---
    int p1[8], p2[4];
    if (MODE == 0) {
        #pragma unroll
        for (int j = 0; j < 8; ++j) p1[j] = perm[s1 * 8 + j];
        p2[0] = perm[s1 * 8 + 8 + c2];
        p2[1] = perm[s1 * 8 + 8 + c2 + 1];
        p2[2] = perm[s1 * 8 + 8 + c2 + 4];
        p2[3] = perm[s1 * 8 + 8 + c2 + 5];
    }

    #pragma unroll
    for (int tt = 0; tt < 4; ++tt) {
        const int tile = blockIdx.y * 32 + waveId * 4 + tt;
        const int b0   = tile * 16;
        const int b    = b0 + ln;     // both wave-halves load the same row

        float x1[8], x2s[4];
        if (MODE == 0) {
            const float* row = src + (size_t)b * 4096;
            #pragma unroll
            for (int j = 0; j < 8; ++j) x1[j] = row[p1[j]];
            #pragma unroll
            for (int j = 0; j < 4; ++j) x2s[j] = row[p2[j]];
        } else {
            const float* base = src + (size_t)b * 4096 + s1 * 8;
            v4f a0 = *(const v4f*)(base);
            v4f a1 = *(const v4f*)(base + 4);
            x1[0] = a0.x; x1[1] = a0.y; x1[2] = a0.z; x1[3] = a0.w;
            x1[4] = a1.x; x1[5] = a1.y; x1[6] = a1.z; x1[7] = a1.w;
            float2 t0 = *(const float2*)(base + 8  + c2);
            float2 t1 = *(const float2*)(base + 12 + c2);
            x2s[0] = t0.x; x2s[1] = t0.y; x2s[2] = t1.x; x2s[3] = t1.y;
        }

        // --- K=64 as 16 chained f32 WMMAs; A built with 2 v_mul per step ----
        v8f acc = {};
        #pragma unroll
        for (int s = 0; s < 16; ++s) {
            v2f A;
            float xi = x1[s >> 1];
            A.x = xi * x2s[(s & 1) * 2 + 0];   // O[row, 4s + 2*hi]
            A.y = xi * x2s[(s & 1) * 2 + 1];   // O[row, 4s + 2*hi + 1]
            acc = __builtin_amdgcn_wmma_f32_16x16x4_f32(
                false, A, false, Bm[s], (short)0, acc, false, false);
        }

        // --- store: lane owns column N=ln, rows M = v + 8*hi ----------------
        float* dp = dst + (size_t)(b0 + 8 * hi) * 4096 + fout;
        #pragma unroll
        for (int v = 0; v < 8; ++v) {
            float val = acc[v];
            if (MODE == 2) val = val * pd + bs;
            dp[(size_t)v * 4096] = val;
        }
    }
}

// ---------------------------------------------------------------------------
extern "C" void kernel_launch(void* const* d_in, const int* in_sizes, int n_in,
                              void* d_out, int out_size, void* d_ws, size_t ws_size,
                              hipStream_t stream)
{
    (void)in_sizes; (void)n_in; (void)out_size; (void)ws_size;
    const float* x     = (const float*)d_in[0];
    const float* g0    = (const float*)d_in[1];
    const float* g1    = (const float*)d_in[2];
    const float* g2    = (const float*)d_in[3];
    const float* alpha = (const float*)d_in[4];
    const float* pds   = (const float*)d_in[5];
    const float* bias  = (const float*)d_in[6];
    const int*   perm  = (const int*)d_in[7];
    float* out = (float*)d_out;

    char* ws = (char*)d_ws;
    float* h  = (float*)ws;                                  // 64 MB: (4096,512,8)
    float* W0 = (float*)(ws + (size_t)64 * 1024 * 1024);     // 1 MB each
    float* W1 = (float*)(ws + (size_t)65 * 1024 * 1024);
    float* W2 = (float*)(ws + (size_t)66 * 1024 * 1024);

    gate_reduce_kernel<<<(256 * 1024) / 256,       256, 0, stream>>>(g0, W0, 256);
    gate_reduce_kernel<<<(255 * 1024 + 255) / 256, 256, 0, stream>>>(g1, W1, 255);
    gate_reduce_kernel<<<(256 * 1024) / 256,       256, 0, stream>>>(g2, W2, 256);

    dim3 blk(256);
    layer_kernel<0><<<dim3(256, 8), blk, 0, stream>>>(x, h,   W0, perm,    nullptr, nullptr, nullptr, 0);
    layer_kernel<1><<<dim3(255, 8), blk, 0, stream>>>(h, h,   W1, nullptr, nullptr, nullptr, nullptr, 1);
    layer_kernel<2><<<dim3(256, 8), blk, 0, stream>>>(h, out, W2, nullptr, pds,     bias,    alpha,   0);
}